// LaplacianBlender_69930657513807
// MI455X (gfx1250) — compile-verified
//
#include <hip/hip_runtime.h>
#include <math.h>

// Laplacian-stack blend, 45-tap separable Gaussian per level, fp32 WMMA
// (V_WMMA_F32_16X16X4_F32) Toeplitz-matmul formulation of the 1D convs.
// Round 4: fix async-to-LDS builtin pointer types (v2i addrspace-qualified).

#define KS    45      // kernel taps
#define PADK  22      // KS/2
#define NIMG  16
#define CH    3
#define HH    512
#define WW    512
#define TILES_X (WW/16)
#define TILES_Y (HH/16)
#define BLUR_LEVELS 4 // levels 0..3 blur; level 4 is plain lerp

typedef float v2f __attribute__((ext_vector_type(2)));
typedef float v8f __attribute__((ext_vector_type(8)));

#if defined(__has_builtin)
#if __has_builtin(__builtin_amdgcn_global_load_async_to_lds_b64)
#define HAVE_ASYNC_LDS 1
#endif
#endif

#ifdef HAVE_ASYNC_LDS
// Matches clang's parameter type: 'int __vector(2) __device__ *' (addrspace 1)
// and the LDS-side pointer in addrspace 3.
typedef int v2i_vs __attribute__((vector_size(8)));
typedef __attribute__((address_space(1))) v2i_vs* async_gptr;
typedef __attribute__((address_space(3))) v2i_vs* async_lptr;
#endif

// ---------------------------------------------------------------------------
// Normalized Gaussian taps for sigma = 2^lvl (lvl = 0..3), float64 generation
// like the reference.
// ---------------------------------------------------------------------------
__global__ void lapblend_init_taps(float* __restrict__ taps) {
    int lvl = threadIdx.x;
    if (lvl < BLUR_LEVELS) {
        double sigma = 1.0;
        for (int i = 0; i < lvl; ++i) sigma *= 2.0;
        double v[KS];
        double s = 0.0;
        for (int t = 0; t < KS; ++t) {
            double x = (double)t - (double)PADK;
            v[t] = exp(-(x * x) / (2.0 * sigma * sigma));
            s += v[t];
        }
        for (int t = 0; t < KS; ++t)
            taps[lvl * KS + t] = (float)(v[t] / s);
    }
}

// ---------------------------------------------------------------------------
// Vertical 45-tap conv with replicate padding.
// One wave per 16x16 output tile:  D(16x16) = T(16x64) x In(64x16)
//   T[m][r] = g[r-m]  (banded Toeplitz, zero outside [0,45))  -> A operand
//   In[r][n] = src[clamp(y0-22+r)][x0+n]                      -> B operand
// Interior tiles use one per-lane base address + compile-time imm offsets.
// ---------------------------------------------------------------------------
__global__ __launch_bounds__(256)
void lapblend_blur_v(const float* __restrict__ src, float* __restrict__ dst,
                     const float* __restrict__ taps) {
    const int lane = threadIdx.x & 31;
    const int wid  = threadIdx.x >> 5;
    const int tile = blockIdx.x * 8 + wid;          // grid is exact
    const int plane = tile / (TILES_Y * TILES_X);
    const int rem   = tile % (TILES_Y * TILES_X);
    const int ty = rem / TILES_X;
    const int y0 = ty * 16;
    const int x0 = (rem % TILES_X) * 16;

    const int m  = lane & 15;   // A-row / B-col / D-col index for this lane
    const int g2 = lane >> 4;   // K half-select (lanes 16-31 carry K+2)

    const float* base = src + (size_t)plane * (HH * WW);

    // A operand: Toeplitz of the taps, built once into registers.
    v2f a[16];
#pragma unroll
    for (int kb = 0; kb < 16; ++kb) {
        int r0 = kb * 4 + g2 * 2;
        int i0 = r0 - m, i1 = r0 + 1 - m;
        a[kb].x = (i0 >= 0 && i0 < KS) ? taps[i0] : 0.0f;
        a[kb].y = (i1 >= 0 && i1 < KS) ? taps[i1] : 0.0f;
    }

    v8f c = {};
    if (ty >= 2 && ty <= TILES_Y - 3) {
        // Fast path: rows y0-22 .. y0+41 all in range; one base + imm offsets.
        const float* lb = base + (size_t)(y0 - PADK + 2 * g2) * WW + x0 + m;
        v2f b[16];
#pragma unroll
        for (int kb = 0; kb < 16; ++kb) {
            b[kb].x = lb[kb * 4 * WW];
            b[kb].y = lb[kb * 4 * WW + WW];
        }
#pragma unroll
        for (int kb = 0; kb < 16; ++kb)
            c = __builtin_amdgcn_wmma_f32_16x16x4_f32(false, a[kb], false, b[kb],
                                                      (short)0, c, false, false);
    } else {
        // Border path: replicate-pad via index clamping.
#pragma unroll
        for (int kb = 0; kb < 16; ++kb) {
            int r0  = kb * 4 + g2 * 2;
            int yy0 = y0 - PADK + r0;
            int yy1 = yy0 + 1;
            yy0 = yy0 < 0 ? 0 : (yy0 > HH - 1 ? HH - 1 : yy0);
            yy1 = yy1 < 0 ? 0 : (yy1 > HH - 1 ? HH - 1 : yy1);
            v2f b;
            b.x = base[(size_t)yy0 * WW + x0 + m];
            b.y = base[(size_t)yy1 * WW + x0 + m];
            c = __builtin_amdgcn_wmma_f32_16x16x4_f32(false, a[kb], false, b,
                                                      (short)0, c, false, false);
        }
    }

    float* obase = dst + (size_t)plane * (HH * WW) + (size_t)(y0 + 8 * g2) * WW + x0 + m;
#pragma unroll
    for (int j = 0; j < 8; ++j)
        obase[(size_t)j * WW] = c[j];
}

// ---------------------------------------------------------------------------
// Horizontal 45-tap conv with replicate padding.
// One wave per 16x16 tile:  D(16x16) = A(16x64) x T'(64x16)
//   A[m][k]  = src[y0+m][clamp(x0-22+k)]  staged through LDS (pitch 68 ->
//              conflict-free column reads, 8B aligned for ds_load_b64)
//   T'[k][n] = g[k-n]                                         -> B operand
// Interior tiles stage via GLOBAL_LOAD_ASYNC_TO_LDS_B64 when available.
// ---------------------------------------------------------------------------
__global__ __launch_bounds__(256)
void lapblend_blur_h(const float* __restrict__ src, float* __restrict__ dst,
                     const float* __restrict__ taps) {
    __shared__ float lds[8][16 * 68];
    const int lane = threadIdx.x & 31;
    const int wid  = threadIdx.x >> 5;
    float* tl = lds[wid];

    const int tile  = blockIdx.x * 8 + wid;         // grid is exact
    const int plane = tile / (TILES_Y * TILES_X);
    const int rem   = tile % (TILES_Y * TILES_X);
    const int ty = rem / TILES_X;
    const int tx = rem % TILES_X;
    const int y0 = ty * 16, x0 = tx * 16;

    const int m  = lane & 15;
    const int g2 = lane >> 4;

    const float* base = src + (size_t)plane * (HH * WW);

    if (tx >= 2 && tx <= TILES_X - 3) {
        // Fast path: cols x0-22 .. x0+41 all in range. Each wave covers a
        // whole 64-float row per iteration with b64 transfers (8B aligned:
        // x0-22 is even).
        const float* lb = base + (size_t)y0 * WW + (x0 - PADK) + 2 * lane;
        float* ld = tl + 2 * lane;
#ifdef HAVE_ASYNC_LDS
#pragma unroll
        for (int r = 0; r < 16; ++r) {
            __builtin_amdgcn_global_load_async_to_lds_b64(
                (async_gptr)(lb + (size_t)r * WW), (async_lptr)(ld + r * 68),
                /*offset=*/0, /*cpol=*/0);
        }
#if __has_builtin(__builtin_amdgcn_s_wait_asynccnt)
        __builtin_amdgcn_s_wait_asynccnt(0);
#else
        asm volatile("s_wait_asynccnt 0x0" ::: "memory");
#endif
#else
#pragma unroll
        for (int r = 0; r < 16; ++r) {
            v2f d = *(const v2f*)(lb + (size_t)r * WW);
            *(v2f*)(ld + r * 68) = d;
        }
#endif
    } else {
        // Border path: scalar with x clamping.
#pragma unroll
        for (int i = 0; i < 32; ++i) {
            int idx = lane + 32 * i;
            int row = idx >> 6, col = idx & 63;
            int xx = x0 - PADK + col;
            xx = xx < 0 ? 0 : (xx > WW - 1 ? WW - 1 : xx);
            tl[row * 68 + col] = base[(size_t)(y0 + row) * WW + xx];
        }
    }
    __syncthreads();

    // B operand: Toeplitz taps in registers.
    v2f bt[16];
#pragma unroll
    for (int kb = 0; kb < 16; ++kb) {
        int c0 = kb * 4 + 2 * g2;
        int i0 = c0 - m, i1 = c0 + 1 - m;
        bt[kb].x = (i0 >= 0 && i0 < KS) ? taps[i0] : 0.0f;
        bt[kb].y = (i1 >= 0 && i1 < KS) ? taps[i1] : 0.0f;
    }

    v8f c = {};
#pragma unroll
    for (int kb = 0; kb < 16; ++kb) {
        int c0 = kb * 4 + 2 * g2;
        v2f av = *(const v2f*)(tl + m * 68 + c0);   // ds_load_b64, no conflicts
        c = __builtin_amdgcn_wmma_f32_16x16x4_f32(false, av, false, bt[kb],
                                                  (short)0, c, false, false);
    }

    float* obase = dst + (size_t)plane * (HH * WW) + (size_t)(y0 + 8 * g2) * WW + x0 + m;
#pragma unroll
    for (int j = 0; j < 8; ++j)
        obase[(size_t)j * WW] = c[j];
}

// ---------------------------------------------------------------------------
// acc (+)= lap0 + m*(lap1-lap0), mask broadcast over channels. float4 form.
// ---------------------------------------------------------------------------
__global__ void lapblend_blend(const float4* __restrict__ cur0,
                               const float4* __restrict__ cur1,
                               const float4* __restrict__ curm,
                               const float4* __restrict__ b0,
                               const float4* __restrict__ b1,
                               float4* __restrict__ acc, int initFlag) {
    const size_t HWq  = (size_t)HH * WW / 4;
    const size_t PLq  = (size_t)CH * HWq;
    size_t q = (size_t)blockIdx.x * blockDim.x + threadIdx.x;
    if (q >= (size_t)NIMG * PLq) return;
    size_t img = q / PLq;
    size_t hwq = q % HWq;
    float4 mv = curm[img * HWq + hwq];
    float4 c0 = cur0[q], c1 = cur1[q], v0 = b0[q], v1 = b1[q];
    float4 r;
    float l0, l1;
    l0 = c0.x - v0.x; l1 = c1.x - v1.x; r.x = l0 + mv.x * (l1 - l0);
    l0 = c0.y - v0.y; l1 = c1.y - v1.y; r.y = l0 + mv.y * (l1 - l0);
    l0 = c0.z - v0.z; l1 = c1.z - v1.z; r.z = l0 + mv.z * (l1 - l0);
    l0 = c0.w - v0.w; l1 = c1.w - v1.w; r.w = l0 + mv.w * (l1 - l0);
    if (initFlag) {
        acc[q] = r;
    } else {
        float4 a = acc[q];
        a.x += r.x; a.y += r.y; a.z += r.z; a.w += r.w;
        acc[q] = a;
    }
}

// Last level: acc += cur0 + m*(cur1-cur0). float4 form.
__global__ void lapblend_final(const float4* __restrict__ cur0,
                               const float4* __restrict__ cur1,
                               const float4* __restrict__ curm,
                               float4* __restrict__ acc) {
    const size_t HWq  = (size_t)HH * WW / 4;
    const size_t PLq  = (size_t)CH * HWq;
    size_t q = (size_t)blockIdx.x * blockDim.x + threadIdx.x;
    if (q >= (size_t)NIMG * PLq) return;
    size_t img = q / PLq;
    size_t hwq = q % HWq;
    float4 mv = curm[img * HWq + hwq];
    float4 c0 = cur0[q], c1 = cur1[q];
    float4 a = acc[q];
    a.x += c0.x + mv.x * (c1.x - c0.x);
    a.y += c0.y + mv.y * (c1.y - c0.y);
    a.z += c0.z + mv.z * (c1.z - c0.z);
    a.w += c0.w + mv.w * (c1.w - c0.w);
    acc[q] = a;
}

// ---------------------------------------------------------------------------
extern "C" void kernel_launch(void* const* d_in, const int* in_sizes, int n_in,
                              void* d_out, int out_size, void* d_ws, size_t ws_size,
                              hipStream_t stream) {
    (void)in_sizes; (void)n_in; (void)out_size; (void)ws_size;

    const float* in0 = (const float*)d_in[0];   // (16,3,512,512)
    const float* in1 = (const float*)d_in[1];   // (16,3,512,512)
    const float* inm = (const float*)d_in[2];   // (16,1,512,512)
    float* out = (float*)d_out;
    float* ws  = (float*)d_ws;

    const size_t P3 = (size_t)NIMG * CH * HH * WW;  // 12,582,912 floats
    const size_t P1 = (size_t)NIMG * 1  * HH * WW;  //  4,194,304 floats

    // Workspace layout (floats): taps | setA(0,1,m) | setB(0,1,m) | tmp
    float* taps = ws;
    float* A0 = ws + 256;
    float* A1 = A0 + P3;
    float* Am = A1 + P3;
    float* B0 = Am + P1;
    float* B1 = B0 + P3;
    float* Bm = B1 + P3;
    float* TMP = Bm + P1;   // total ~71.3M floats (~285 MB)

    lapblend_init_taps<<<1, 64, 0, stream>>>(taps);

    const int tileBlocks3 = (NIMG * CH * TILES_Y * TILES_X) / 8; // 6144
    const int tileBlocks1 = (NIMG * 1  * TILES_Y * TILES_X) / 8; // 2048
    const int ewBlocks    = (int)((P3 / 4 + 255) / 256);         // 12288

    const float *c0 = in0, *c1 = in1, *cm = inm;
    float* setA[3] = {A0, A1, Am};
    float* setB[3] = {B0, B1, Bm};
    float** dst   = setA;
    float** other = setB;

    for (int lvl = 0; lvl < BLUR_LEVELS; ++lvl) {
        const float* tp = taps + lvl * KS;

        lapblend_blur_v<<<tileBlocks3, 256, 0, stream>>>(c0, TMP, tp);
        lapblend_blur_h<<<tileBlocks3, 256, 0, stream>>>(TMP, dst[0], tp);

        lapblend_blur_v<<<tileBlocks3, 256, 0, stream>>>(c1, TMP, tp);
        lapblend_blur_h<<<tileBlocks3, 256, 0, stream>>>(TMP, dst[1], tp);

        lapblend_blur_v<<<tileBlocks1, 256, 0, stream>>>(cm, TMP, tp);
        lapblend_blur_h<<<tileBlocks1, 256, 0, stream>>>(TMP, dst[2], tp);

        lapblend_blend<<<ewBlocks, 256, 0, stream>>>(
            (const float4*)c0, (const float4*)c1, (const float4*)cm,
            (const float4*)dst[0], (const float4*)dst[1],
            (float4*)out, lvl == 0 ? 1 : 0);
        c0 = dst[0]; c1 = dst[1]; cm = dst[2];
        float** t = dst; dst = other; other = t;
    }

    lapblend_final<<<ewBlocks, 256, 0, stream>>>(
        (const float4*)c0, (const float4*)c1, (const float4*)cm, (float4*)out);
}